// DCTHFClip_54786602828027
// MI455X (gfx1250) — compile-verified
//
#include <hip/hip_runtime.h>

// ---------------------------------------------------------------------------
// DCT high-freq clip + reconstruct for MI455X (gfx1250, wave32, WMMA)
//   x: (64, 576, 1024) f32
//   x_dct = C576 @ x          (per batch)         -- f32 WMMA 16x16x4
//   L from quantile(|C576 @ mean_{b,d}(x)|, 0.7)  -- device-computed
//   out0 = (ClT @ x_dct[:, :L]) as f16, out1 = x_dct[:, :L] f32
// Data movement: GLOBAL_LOAD_ASYNC_TO_LDS_B128 + double-buffered LDS tiles,
// fenced with s_wait_asynccnt + workgroup barrier (ASYNCcnt protocol).
// ---------------------------------------------------------------------------

#define B_  64
#define S_  576
#define D_  1024
#define PI_ 3.14159265358979323846f

typedef __attribute__((ext_vector_type(2))) float v2f;
typedef __attribute__((ext_vector_type(8))) float v8f;

// --- gfx1250 async copy helpers (inline asm: portable across toolchains) ---
__device__ __forceinline__ unsigned lds_addr(const void* p) {
    // generic pointer to LDS: addr[31:0] is the LDS byte offset
    return (unsigned)(unsigned long long)p;
}
__device__ __forceinline__ void async_b128(unsigned lds, unsigned goff,
                                           const void* sbase) {
    asm volatile("global_load_async_to_lds_b128 %0, %1, %2"
                 :: "v"(lds), "v"(goff), "s"(sbase) : "memory");
}
__device__ __forceinline__ void wait_async0() {
    asm volatile("s_wait_asynccnt 0x0" ::: "memory");
}

// ---------------- DCT-576 matrix precompute (row-major, stride 576) --------
__global__ void build_dct576(float* __restrict__ Cmat) {
    int idx = blockIdx.x * blockDim.x + threadIdx.x;
    if (idx >= S_ * S_) return;
    int k = idx / S_, s = idx % S_;
    float scale = (k == 0) ? sqrtf(1.0f / S_) : sqrtf(2.0f / S_);
    Cmat[idx] = cosf((PI_ / S_) * ((float)s + 0.5f) * (float)k) * scale;
}

// ---------------- xbar[s] = mean over (b,d) of x[b,s,d] --------------------
__global__ __launch_bounds__(256) void mean_bd(const float* __restrict__ x,
                                               float* __restrict__ xbar) {
    int s = blockIdx.x;
    float acc = 0.f;
    for (int b = 0; b < B_; ++b) {
        const float* p = x + ((size_t)b * S_ + s) * D_;
        for (int d = threadIdx.x; d < D_; d += 256) acc += p[d];
    }
    __shared__ float red[256];
    red[threadIdx.x] = acc;
    __syncthreads();
    for (int off = 128; off > 0; off >>= 1) {
        if ((int)threadIdx.x < off) red[threadIdx.x] += red[threadIdx.x + off];
        __syncthreads();
    }
    if (threadIdx.x == 0) xbar[s] = red[0] / (float)(B_ * D_);
}

// ---------------- m[k]=|C@xbar|, threshold=quantile0.7, L=last idx > thr ---
__global__ void select_L(const float* __restrict__ Cmat,
                         const float* __restrict__ xbar,
                         int* __restrict__ Lp) {
    __shared__ float sm[S_];
    __shared__ float q[2];
    __shared__ int   last;
    int k = threadIdx.x;
    float acc = 0.f;
    for (int s = 0; s < S_; ++s) acc += Cmat[(size_t)k * S_ + s] * xbar[s];
    sm[k] = fabsf(acc);
    if (k == 0) last = -1;
    __syncthreads();
    float mk = sm[k];
    int rank = 0;                          // strict total order (ties by index)
    for (int j = 0; j < S_; ++j) {
        float mj = sm[j];
        rank += (mj < mk) || (mj == mk && j < k);
    }
    if (rank == 402) q[0] = mk;            // pos = 0.7*575 = 402.5 -> midpoint
    if (rank == 403) q[1] = mk;
    __syncthreads();
    float thr = 0.5f * (q[0] + q[1]);
    if (mk > thr) atomicMax(&last, k);
    __syncthreads();
    if (k == 0) Lp[0] = last;              // keep rows [0, last)
}

// ---------------- ClT[l][k] = Cl[k][l], row stride LP = align4(L) ----------
__global__ void build_dctL(const int* __restrict__ Lp, float* __restrict__ ClT) {
    int L = *Lp;
    if (L <= 0) return;
    int LP = (L + 3) & ~3;                 // 16B-aligned rows for async B128
    int idx = blockIdx.x * blockDim.x + threadIdx.x;
    int l = idx / LP, k = idx % LP;
    if (l >= L || k >= L) return;
    float scale = (k == 0) ? sqrtf(1.0f / (float)L) : sqrtf(2.0f / (float)L);
    ClT[(size_t)l * LP + k] =
        cosf((PI_ / (float)L) * ((float)l + 0.5f) * (float)k) * scale;
}

// ---------------- GEMM1: x_dct_trunc[b,k,d] = sum_s C[k,s] x[b,s,d], k<L ---
// 256 thr (8 waves), tile 64(M) x 128(N), K chunks of 16, async double buffer.
__global__ __launch_bounds__(256) void dct_gemm1(const float* __restrict__ x,
                                                 const float* __restrict__ Cmat,
                                                 const int* __restrict__ Lp,
                                                 void* __restrict__ out_base) {
    const int L = *Lp;
    if (L <= 0) return;
    const int k0 = blockIdx.y * 64;
    if (k0 >= L) return;                   // block-uniform: EXEC stays full
    const int d0 = blockIdx.x * 128;
    const int b  = blockIdx.z;
    float* xt = (float*)((char*)out_base + (size_t)B_ * L * D_ * 2);
    const float* xb = x + (size_t)b * S_ * D_;

    __shared__ float As[2][64][20];        // 80B rows: 16B aligned, no conflicts
    __shared__ float Bs[2][16][136];       // 544B rows: 16B aligned, no conflicts

    const int tid  = threadIdx.x;
    const int lane = tid & 31, wave = tid >> 5;
    const int wk = wave >> 1, wd = wave & 1;
    const int lh = lane >> 4, ll = lane & 15;

    // async staging map: A 64x16 -> 1 b128/thread, B 16x128 -> 2 b128/thread
    const int ar = tid >> 2, ac = (tid & 3) << 2;
    const int gk = k0 + ar;
    if (gk >= L) {                          // zero pad rows once (persist)
        #pragma unroll
        for (int z = 0; z < 4; ++z) { As[0][ar][ac + z] = 0.f; As[1][ar][ac + z] = 0.f; }
    }

    v8f acc[4] = {};

    // prologue: stage chunk 0 into buffer 0
    if (gk < L)
        async_b128(lds_addr(&As[0][ar][ac]), (unsigned)((gk * S_ + ac) * 4), Cmat);
    #pragma unroll
    for (int j = 0; j < 2; ++j) {
        int lin = (tid << 3) + (j << 2);
        int r = lin >> 7, c = lin & 127;
        async_b128(lds_addr(&Bs[0][r][c]), (unsigned)((r * D_ + d0 + c) * 4), xb);
    }

    for (int i = 0; i < S_ / 16; ++i) {
        const int cur = i & 1;
        wait_async0();                      // my chunk-i transfers landed
        __syncthreads();                    // everyone's landed; buf cur^1 free
        if (i + 1 < S_ / 16) {              // prefetch chunk i+1
            const int s0 = (i + 1) << 4;
            if (gk < L)
                async_b128(lds_addr(&As[cur ^ 1][ar][ac]),
                           (unsigned)((gk * S_ + s0 + ac) * 4), Cmat);
            #pragma unroll
            for (int j = 0; j < 2; ++j) {
                int lin = (tid << 3) + (j << 2);
                int r = lin >> 7, c = lin & 127;
                async_b128(lds_addr(&Bs[cur ^ 1][r][c]),
                           (unsigned)(((s0 + r) * D_ + d0 + c) * 4), xb);
            }
        }
        #pragma unroll
        for (int kk = 0; kk < 4; ++kk) {    // 16 WMMA per chunk per wave
            int kb = kk * 4 + (lh << 1);
            v2f a;
            a.x = As[cur][(wk << 4) + ll][kb];
            a.y = As[cur][(wk << 4) + ll][kb + 1];
            #pragma unroll
            for (int n = 0; n < 4; ++n) {
                int bn = (wd << 6) + (n << 4) + ll;
                v2f bb;
                bb.x = Bs[cur][kb][bn];
                bb.y = Bs[cur][kb + 1][bn];
                acc[n] = __builtin_amdgcn_wmma_f32_16x16x4_f32(
                    false, a, false, bb, (short)0, acc[n], false, false);
            }
        }
    }
    // store (C/D layout: row = j + 8*lh, col = ll)
    const int rbase = k0 + (wk << 4) + (lh << 3);
    const int cbase = d0 + (wd << 6) + ll;
    #pragma unroll
    for (int n = 0; n < 4; ++n)
        #pragma unroll
        for (int j = 0; j < 8; ++j) {
            int r = rbase + j;
            if (r < L)
                xt[((size_t)b * L + r) * D_ + cbase + (n << 4)] = acc[n][j];
        }
}

// ---------------- GEMM2: recon[b,l,d] = sum_{k<L} ClT[l,k] xt[b,k,d] -> f16
__global__ __launch_bounds__(256) void dct_gemm2(const int* __restrict__ Lp,
                                                 const float* __restrict__ ClT,
                                                 void* __restrict__ out_base) {
    const int L = *Lp;
    if (L <= 0) return;
    const int l0 = blockIdx.y * 64;
    if (l0 >= L) return;
    const int LP = (L + 3) & ~3;
    const int d0 = blockIdx.x * 128;
    const int b  = blockIdx.z;
    const float* xt  = (const float*)((const char*)out_base + (size_t)B_ * L * D_ * 2);
    const float* xtb = xt + (size_t)b * L * D_;
    _Float16*    rec = (_Float16*)out_base;

    __shared__ float As[2][64][20];
    __shared__ float Bs[2][16][136];

    const int tid  = threadIdx.x;
    const int lane = tid & 31, wave = tid >> 5;
    const int wk = wave >> 1, wd = wave & 1;
    const int lh = lane >> 4, ll = lane & 15;

    const int NF = L >> 4, T = L & 15;      // full chunks + tail rows

    const int ar = tid >> 2, ac = (tid & 3) << 2;
    const int gl = l0 + ar;
    if (gl >= L) {
        #pragma unroll
        for (int z = 0; z < 4; ++z) { As[0][ar][ac + z] = 0.f; As[1][ar][ac + z] = 0.f; }
    }

    v8f acc[4] = {};

    if (NF > 0) {                           // prologue: chunk 0
        if (gl < L)
            async_b128(lds_addr(&As[0][ar][ac]), (unsigned)((gl * LP + ac) * 4), ClT);
        #pragma unroll
        for (int j = 0; j < 2; ++j) {
            int lin = (tid << 3) + (j << 2);
            int r = lin >> 7, c = lin & 127;
            async_b128(lds_addr(&Bs[0][r][c]), (unsigned)((r * D_ + d0 + c) * 4), xtb);
        }
    }
    for (int i = 0; i < NF; ++i) {
        const int cur = i & 1;
        wait_async0();
        __syncthreads();
        if (i + 1 < NF) {
            const int c0 = (i + 1) << 4;
            if (gl < L)
                async_b128(lds_addr(&As[cur ^ 1][ar][ac]),
                           (unsigned)((gl * LP + c0 + ac) * 4), ClT);
            #pragma unroll
            for (int j = 0; j < 2; ++j) {
                int lin = (tid << 3) + (j << 2);
                int r = lin >> 7, c = lin & 127;
                async_b128(lds_addr(&Bs[cur ^ 1][r][c]),
                           (unsigned)(((c0 + r) * D_ + d0 + c) * 4), xtb);
            }
        }
        #pragma unroll
        for (int kk = 0; kk < 4; ++kk) {
            int kb = kk * 4 + (lh << 1);
            v2f a;
            a.x = As[cur][(wk << 4) + ll][kb];
            a.y = As[cur][(wk << 4) + ll][kb + 1];
            #pragma unroll
            for (int n = 0; n < 4; ++n) {
                int bn = (wd << 6) + (n << 4) + ll;
                v2f bb;
                bb.x = Bs[cur][kb][bn];
                bb.y = Bs[cur][kb + 1][bn];
                acc[n] = __builtin_amdgcn_wmma_f32_16x16x4_f32(
                    false, a, false, bb, (short)0, acc[n], false, false);
            }
        }
    }
    if (T > 0) {                            // tail chunk: synchronous, zero-padded
        __syncthreads();
        const int c0 = NF << 4;
        #pragma unroll
        for (int i2 = 0; i2 < 4; ++i2) {
            int idx = tid + i2 * 256;
            int r = idx >> 4, c = idx & 15;
            int rl = l0 + r, gc = c0 + c;
            As[0][r][c] = (rl < L && gc < L) ? ClT[(size_t)rl * LP + gc] : 0.f;
        }
        #pragma unroll
        for (int i2 = 0; i2 < 8; ++i2) {
            int idx = tid + i2 * 256;
            int r = idx >> 7, c = idx & 127;
            int gkk = c0 + r;
            Bs[0][r][c] = (gkk < L) ? xtb[(size_t)gkk * D_ + d0 + c] : 0.f;
        }
        __syncthreads();
        #pragma unroll
        for (int kk = 0; kk < 4; ++kk) {
            int kb = kk * 4 + (lh << 1);
            v2f a;
            a.x = As[0][(wk << 4) + ll][kb];
            a.y = As[0][(wk << 4) + ll][kb + 1];
            #pragma unroll
            for (int n = 0; n < 4; ++n) {
                int bn = (wd << 6) + (n << 4) + ll;
                v2f bb;
                bb.x = Bs[0][kb][bn];
                bb.y = Bs[0][kb + 1][bn];
                acc[n] = __builtin_amdgcn_wmma_f32_16x16x4_f32(
                    false, a, false, bb, (short)0, acc[n], false, false);
            }
        }
    }
    const int rbase = l0 + (wk << 4) + (lh << 3);
    const int cbase = d0 + (wd << 6) + ll;
    #pragma unroll
    for (int n = 0; n < 4; ++n)
        #pragma unroll
        for (int j = 0; j < 8; ++j) {
            int r = rbase + j;
            if (r < L)
                rec[((size_t)b * L + r) * D_ + cbase + (n << 4)] =
                    (_Float16)acc[n][j];
        }
}

// ---------------------------------------------------------------------------
extern "C" void kernel_launch(void* const* d_in, const int* in_sizes, int n_in,
                              void* d_out, int out_size, void* d_ws, size_t ws_size,
                              hipStream_t stream) {
    const float* x = (const float*)d_in[0];

    // workspace (floats): C576 [576*576] | ClT [576*580 padded] | xbar | L
    float* ws   = (float*)d_ws;
    float* Cmat = ws;
    float* ClT  = ws + (size_t)S_ * S_;
    float* xbar = ClT + (size_t)S_ * 580;
    int*   Lp   = (int*)(xbar + S_);

    build_dct576<<<(S_ * S_ + 255) / 256, 256, 0, stream>>>(Cmat);
    mean_bd<<<S_, 256, 0, stream>>>(x, xbar);
    select_L<<<1, S_, 0, stream>>>(Cmat, xbar, Lp);
    build_dctL<<<(S_ * 580 + 255) / 256, 256, 0, stream>>>(Lp, ClT);

    dim3 grid(D_ / 128, (S_ + 63) / 64, B_);   // blocks self-disable vs L
    dct_gemm1<<<grid, 256, 0, stream>>>(x, Cmat, Lp, d_out);
    dct_gemm2<<<grid, 256, 0, stream>>>(Lp, ClT, d_out);
}